// FTUNet_5592047419740
// MI455X (gfx1250) — compile-verified
//
#include <hip/hip_runtime.h>
#include <hip/hip_bf16.h>
#include <math.h>

typedef __attribute__((ext_vector_type(2))) float v2f;
typedef __attribute__((ext_vector_type(8))) float v8f;

#define C_CH   2048
#define HW     4096
#define BATCH  8
#define NU     (C_CH * 3)      // 6144 u-values per batch
#define NSORT  8192            // padded pow2 for bitonic sort

// ---------------------------------------------------------------------------
// Kernel 1: per-(b,c) sum / sumsq / max over 4096 elements.
// One wave32 per (b,c). WMMA f32 16x16x4 with all-ones B as the accumulator:
// D = A*1 + C  =>  sum over all distinct D rows == sum of all A entries.
// Robustly: Sum(all 8 VGPRs over all 32 lanes) = 16 * total (16 identical cols).
// ---------------------------------------------------------------------------
__global__ __launch_bounds__(256) void msca_stats_kernel(
    const float* __restrict__ x, float* __restrict__ u)
{
    const int lane = threadIdx.x & 31;
    const int wave = threadIdx.x >> 5;
    const int bc   = blockIdx.x * 8 + wave;           // 2048 blocks * 8 waves = 16384
    const float* p = x + (size_t)bc * HW;

    v2f ones; ones[0] = 1.0f; ones[1] = 1.0f;
    v8f accS = {};   // running sum accumulator (C matrix)
    v8f accQ = {};   // running sum-of-squares accumulator
    float vmax = -INFINITY;

    // 4096 elements / (128 per iter) = 32 iterations.
    // Each lane loads one float4 (wave covers 512B contiguous), feeds 2 A-tiles.
    for (int it = 0; it < HW / 128; ++it) {
        const float4 d = *reinterpret_cast<const float4*>(p + it * 128 + lane * 4);

        v2f a0;  a0[0] = d.x;        a0[1] = d.y;
        v2f a1;  a1[0] = d.z;        a1[1] = d.w;
        v2f q0;  q0[0] = d.x * d.x;  q0[1] = d.y * d.y;
        v2f q1;  q1[0] = d.z * d.z;  q1[1] = d.w * d.w;

        vmax = fmaxf(vmax, fmaxf(fmaxf(d.x, d.y), fmaxf(d.z, d.w)));

        accS = __builtin_amdgcn_wmma_f32_16x16x4_f32(false, a0, false, ones, (short)0, accS, false, false);
        accQ = __builtin_amdgcn_wmma_f32_16x16x4_f32(false, q0, false, ones, (short)0, accQ, false, false);
        accS = __builtin_amdgcn_wmma_f32_16x16x4_f32(false, a1, false, ones, (short)0, accS, false, false);
        accQ = __builtin_amdgcn_wmma_f32_16x16x4_f32(false, q1, false, ones, (short)0, accQ, false, false);
    }

    // Per-lane partial over the 8 accumulator registers.
    float s = 0.0f, q = 0.0f;
#pragma unroll
    for (int r = 0; r < 8; ++r) { s += accS[r]; q += accQ[r]; }

    // wave32 cross-lane reduction
#pragma unroll
    for (int off = 16; off > 0; off >>= 1) {
        s    += __shfl_xor(s, off, 32);
        q    += __shfl_xor(q, off, 32);
        vmax  = fmaxf(vmax, __shfl_xor(vmax, off, 32));
    }
    s *= (1.0f / 16.0f);   // 16 replicated columns in D
    q *= (1.0f / 16.0f);

    if (lane == 0) {
        const float n    = (float)HW;
        const float mean = s / n;
        const float var  = (q - n * mean * mean) / (n - 1.0f);   // unbiased
        const float sd   = sqrtf(fmaxf(var, 0.0f));
        float* o = u + (size_t)bc * 3;
        o[0] = mean; o[1] = vmax; o[2] = sd;
    }
}

// ---------------------------------------------------------------------------
// Kernel 2: per-batch median over 6144 u-values (bitonic sort in LDS),
// then cubed centering, grouped conv (k=3), BN(eval), sigmoid -> g[b,c].
// One 256-thread block per batch.
// ---------------------------------------------------------------------------
__global__ __launch_bounds__(256) void msca_gate_kernel(
    const float* __restrict__ u, const float* __restrict__ cfc_w,
    const float* __restrict__ bn_w, const float* __restrict__ bn_b,
    float* __restrict__ g)
{
    __shared__ float smem[NSORT];
    const int b   = blockIdx.x;
    const int tid = threadIdx.x;

    const float* ub = u + (size_t)b * NU;
    for (int i = tid; i < NSORT; i += 256)
        smem[i] = (i < NU) ? ub[i] : INFINITY;
    __syncthreads();

    // bitonic sort, ascending
    for (int k = 2; k <= NSORT; k <<= 1) {
        for (int j = k >> 1; j > 0; j >>= 1) {
            for (int i = tid; i < NSORT; i += 256) {
                const int ixj = i ^ j;
                if (ixj > i) {
                    const float a = smem[i];
                    const float c = smem[ixj];
                    const bool up = ((i & k) == 0);
                    if ((a > c) == up) { smem[i] = c; smem[ixj] = a; }
                }
            }
            __syncthreads();
        }
    }

    const float med = 0.5f * (smem[NU / 2 - 1] + smem[NU / 2]);
    const float bn_scale = 1.0f / sqrtf(1.0f + 1e-5f);

    for (int c = tid; c < C_CH; c += 256) {
        const float* uc = ub + c * 3;
        const float* wc = cfc_w + c * 3;
        float z = 0.0f;
#pragma unroll
        for (int k = 0; k < 3; ++k) {
            const float t = uc[k] - med;
            z += (t * t * t) * wc[k];
        }
        z = z * bn_scale * bn_w[c] + bn_b[c];
        g[(size_t)b * C_CH + c] = 1.0f / (1.0f + expf(-z));
    }
}

// ---------------------------------------------------------------------------
// Kernel 3: out = x * g[bc], float4 streaming. 4096 % 4 == 0 so a float4
// never straddles a channel boundary; bc = (elem index) >> 12 = (vec4) >> 10.
// ---------------------------------------------------------------------------
__global__ __launch_bounds__(256) void msca_scale_kernel(
    const float4* __restrict__ x4, const float* __restrict__ g,
    float4* __restrict__ o4, int total4)
{
    const int i = blockIdx.x * 256 + threadIdx.x;
    if (i >= total4) return;
    const float gg = g[i >> 10];
    float4 v = x4[i];
    v.x *= gg; v.y *= gg; v.z *= gg; v.w *= gg;
    o4[i] = v;
}

// ---------------------------------------------------------------------------
extern "C" void kernel_launch(void* const* d_in, const int* in_sizes, int n_in,
                              void* d_out, int out_size, void* d_ws, size_t ws_size,
                              hipStream_t stream)
{
    const float* x     = (const float*)d_in[0];   // [8,2048,64,64]
    const float* cfc_w = (const float*)d_in[1];   // [2048,3]
    const float* bn_w  = (const float*)d_in[2];   // [2048]
    const float* bn_b  = (const float*)d_in[3];   // [2048]
    float*       out   = (float*)d_out;

    float* u = (float*)d_ws;                       // [8*2048*3] = 49152 floats
    float* g = u + (size_t)BATCH * C_CH * 3;       // [8*2048]   = 16384 floats

    // 1) stats: 16384 (b,c) pairs, one wave each, 8 waves/block
    msca_stats_kernel<<<(BATCH * C_CH) / 8, 256, 0, stream>>>(x, u);

    // 2) median + gate: one block per batch
    msca_gate_kernel<<<BATCH, 256, 0, stream>>>(u, cfc_w, bn_w, bn_b, g);

    // 3) elementwise scale
    const int total4 = (BATCH * C_CH * HW) / 4;    // 16,777,216
    msca_scale_kernel<<<total4 / 256, 256, 0, stream>>>(
        (const float4*)x, g, (float4*)out, total4);
}